// SparsegenAttentionEntityPooler_17557826306585
// MI455X (gfx1250) — compile-verified
//
#include <hip/hip_runtime.h>
#include <math.h>

#define B_ 64
#define T_ 1024
#define H_ 1024
#define O_ 1024
#define LAM_ 0.5f
#define MAXTOK 40

typedef float v2f __attribute__((ext_vector_type(2)));
typedef float v8f __attribute__((ext_vector_type(8)));

// ---------------------------------------------------------------------------
// Kernel 1: one block per batch row.
//  - ent_dot = dot(pooled_entities[b], align_w[:H])        (block reduction)
//  - build list of masked tokens (<=40) from the two spans (thread 0)
//  - per-token score dot(hidden[b,t], align_w[H:])          (wave per token)
//  - sparsegen-lin over the <=40 scores                     (thread 0, serial)
//  - scatter probs, gather pooled[b] = sum p_i * hidden[b,t_i]
// ---------------------------------------------------------------------------
__global__ __launch_bounds__(256) void sparsegen_pool_kernel(
    const float* __restrict__ hidden,          // [B,T,H]
    const int*   __restrict__ token_idxs,      // [1,B,2,2]
    const float* __restrict__ pooled_entities, // [B,H]
    const float* __restrict__ align_w,         // [2H]
    const float* __restrict__ align_b,         // [1]
    float* __restrict__ pooled_ws,             // [B,H]  (workspace)
    float* __restrict__ probs_out)             // [B,T]
{
    const int b    = blockIdx.x;
    const int tid  = threadIdx.x;
    const int lane = tid & 31;
    const int wid  = tid >> 5;

    __shared__ float sWave[8];
    __shared__ float sEntDot;
    __shared__ int   sTok[MAXTOK];
    __shared__ float sZ[MAXTOK];
    __shared__ float sP[MAXTOK];
    __shared__ int   sN;

    // ---- entity-side dot product (whole block) ----
    float part = 0.0f;
    for (int h = tid; h < H_; h += 256)
        part += pooled_entities[b * H_ + h] * align_w[h];
    for (int off = 16; off; off >>= 1)
        part += __shfl_down(part, off, 32);
    if (lane == 0) sWave[wid] = part;
    __syncthreads();

    if (tid == 0) {
        float s = 0.0f;
        for (int i = 0; i < 8; ++i) s += sWave[i];
        sEntDot = s;
        // ---- build masked-token list from the two spans ----
        const int s0 = token_idxs[b * 4 + 0];
        const int e0 = token_idxs[b * 4 + 1];
        const int s1 = token_idxs[b * 4 + 2];
        const int e1 = token_idxs[b * 4 + 3];
        int n = 0;
        for (int t = s0; t < e0 && n < MAXTOK; ++t) sTok[n++] = t;
        for (int t = s1; t < e1 && n < MAXTOK; ++t)
            if (!(t >= s0 && t < e0)) sTok[n++] = t;
        sN = n;
    }
    __syncthreads();
    const int n = sN;

    // ---- per-token hidden-side dot: one wave per token ----
    for (int i = wid; i < n; i += 8) {
        const float* hrow = hidden + (size_t)b * T_ * H_ + (size_t)sTok[i] * H_;
        float acc = 0.0f;
        for (int h = lane; h < H_; h += 32)
            acc += hrow[h] * align_w[H_ + h];
        for (int off = 16; off; off >>= 1)
            acc += __shfl_down(acc, off, 32);
        if (lane == 0) sZ[i] = acc;
    }
    __syncthreads();

    // ---- sparsegen-lin over n entries (serial, n <= 40) ----
    if (tid == 0) {
        const float bias = align_b[0];
        float zmax = -1e30f;
        for (int i = 0; i < n; ++i) {
            sZ[i] += sEntDot + bias;
            zmax = fmaxf(zmax, sZ[i]);
        }
        float z[MAXTOK], srt[MAXTOK];
        for (int i = 0; i < n; ++i) { z[i] = sZ[i] - zmax; srt[i] = z[i]; }
        // insertion sort, descending
        for (int i = 1; i < n; ++i) {
            float v = srt[i];
            int j = i - 1;
            while (j >= 0 && srt[j] < v) { srt[j + 1] = srt[j]; --j; }
            srt[j + 1] = v;
        }
        float cum = 0.0f, kmax = 0.0f, sum_gt = 0.0f;
        for (int i = 0; i < n; ++i) {
            cum += srt[i];
            const float kk = (float)(i + 1);
            if (1.0f - LAM_ + kk * srt[i] > cum) { kmax = kk; sum_gt += srt[i]; }
        }
        const float k   = fmaxf(kmax, 1.0f);
        const float tau = (sum_gt - 1.0f + LAM_) / k;
        for (int i = 0; i < n; ++i)
            sP[i] = fmaxf((z[i] - tau) / (1.0f - LAM_), 0.0f);
    }
    __syncthreads();

    // ---- write probs row: zero then scatter ----
    for (int t = tid; t < T_; t += 256)
        probs_out[b * T_ + t] = 0.0f;
    __syncthreads();
    if (tid < n)
        probs_out[b * T_ + sTok[tid]] = sP[tid];

    // ---- pooled[b,h] = sum_i p_i * hidden[b, t_i, h] ----
    const float* hb = hidden + (size_t)b * T_ * H_;
    for (int h = tid; h < H_; h += 256) {
        float acc = 0.0f;
        for (int i = 0; i < n; ++i)
            acc += sP[i] * hb[(size_t)sTok[i] * H_ + h];
        pooled_ws[b * H_ + h] = acc;
    }
}

// ---------------------------------------------------------------------------
// Kernel 2: projected = tanh(pooled @ out_w + out_b), M=64, N=1024, K=1024.
// One wave per 16x16 tile, V_WMMA_F32_16X16X4_F32 over K in steps of 4.
//   A 16x4 : lanes hold M (lane&15), K = 2*(lane>>4) + {vgpr0,vgpr1}
//   B 4x16 : lanes hold N (lane&15), K split the same way
//   C/D    : M = v + 8*(lane>>4), N = lane&15
// ---------------------------------------------------------------------------
__global__ __launch_bounds__(256) void gemm_tanh_wmma_kernel(
    const float* __restrict__ A,    // pooled [64,1024] row-major
    const float* __restrict__ W,    // out_w  [1024,1024] row-major (K x N)
    const float* __restrict__ bias, // [1024]
    float* __restrict__ out)        // [64,1024]
{
    const int tid  = threadIdx.x;
    const int lane = tid & 31;
    const int wave = blockIdx.x * 8 + (tid >> 5);   // 0..255
    const int tile_m = wave >> 6;                   // 0..3
    const int tile_n = wave & 63;                   // 0..63
    const int half = lane >> 4;
    const int l15  = lane & 15;

    const int arow = tile_m * 16 + l15;
    const int bcol = tile_n * 16 + l15;

    v8f c = {0.f, 0.f, 0.f, 0.f, 0.f, 0.f, 0.f, 0.f};
    for (int k = 0; k < H_; k += 4) {
        const int ka = k + 2 * half;
        v2f a;
        a.x = A[arow * H_ + ka];
        a.y = A[arow * H_ + ka + 1];
        v2f bm;
        bm.x = W[(size_t)ka * O_ + bcol];
        bm.y = W[(size_t)(ka + 1) * O_ + bcol];
        c = __builtin_amdgcn_wmma_f32_16x16x4_f32(
                /*neg_a=*/false, a, /*neg_b=*/false, bm,
                /*c_mod=*/(short)0, c, /*reuse_a=*/false, /*reuse_b=*/false);
    }

    const float bv = bias[bcol];
#pragma unroll
    for (int v = 0; v < 8; ++v) {
        const int row = tile_m * 16 + half * 8 + v;
        out[row * O_ + bcol] = tanhf(c[v] + bv);
    }
}

extern "C" void kernel_launch(void* const* d_in, const int* in_sizes, int n_in,
                              void* d_out, int out_size, void* d_ws, size_t ws_size,
                              hipStream_t stream) {
    const float* hidden          = (const float*)d_in[0]; // [B,T,H]
    const int*   token_idxs      = (const int*)  d_in[1]; // [1,B,2,2]
    const float* pooled_entities = (const float*)d_in[2]; // [B,H]
    const float* align_w         = (const float*)d_in[3]; // [2H,1]
    const float* align_b         = (const float*)d_in[4]; // [1]
    const float* out_w           = (const float*)d_in[5]; // [H,O]
    const float* out_b           = (const float*)d_in[6]; // [O]

    float* projected = (float*)d_out;             // [B,O]
    float* probs     = (float*)d_out + B_ * O_;   // [1,B,T]
    float* pooled_ws = (float*)d_ws;              // [B,H]

    sparsegen_pool_kernel<<<B_, 256, 0, stream>>>(
        hidden, token_idxs, pooled_entities, align_w, align_b, pooled_ws, probs);

    // 256 tiles (4 x 64), 8 waves per block -> 32 blocks
    gemm_tanh_wmma_kernel<<<32, 256, 0, stream>>>(
        pooled_ws, out_w, out_b, projected);
}